// HeteroConvkLayer_90881507983897
// MI455X (gfx1250) — compile-verified
//
#include <hip/hip_runtime.h>
#include <hip/hip_bf16.h>

typedef __attribute__((ext_vector_type(2))) float v2f;
typedef __attribute__((ext_vector_type(8))) float v8f;

// ---------------------------------------------------------------------------
// Copy root features into the concatenated feature matrix F (column block).
// ---------------------------------------------------------------------------
__global__ void copy_root_kernel(const float* __restrict__ x, int f,
                                 float* __restrict__ F, int stride, int colofs,
                                 int n) {
    long long t = (long long)blockIdx.x * blockDim.x + threadIdx.x;
    long long total = (long long)n * f;
    if (t >= total) return;
    int node = (int)(t / f);
    int k = (int)(t % f);
    F[(long long)node * stride + colofs + k] = x[t];
}

// ---------------------------------------------------------------------------
// Build combined weight matrices (W_rel blocks stacked over W_root sums) and
// summed biases. Single block of 256 threads; tiny.
// ---------------------------------------------------------------------------
__global__ void build_weights_kernel(const float* __restrict__ W_rel5,
                                     const float* __restrict__ W_rel12,
                                     const float* __restrict__ b_rel,
                                     const float* __restrict__ W_root5,
                                     const float* __restrict__ W_root12,
                                     float* __restrict__ Wc_t,   // 56x16
                                     float* __restrict__ Wc_d,   // 32x16
                                     float* __restrict__ Wc_v,   // 32x16
                                     float* __restrict__ bias3)  // 3x16
{
    int t = threadIdx.x;
    // tasks: rows 0-4 W_rel5[0]; 5-16 W_rel12[2]; 17-28 W_rel12[4];
    //        29-40 W_rel12[5]; 41-52 sum of W_root12[0,2,4,5]; 53-55 zero
    for (int idx = t; idx < 56 * 16; idx += 256) {
        int r = idx >> 4, h = idx & 15;
        float v;
        if (r < 5)       v = W_rel5[(0 * 5 + r) * 16 + h];
        else if (r < 17) v = W_rel12[(2 * 12 + (r - 5)) * 16 + h];
        else if (r < 29) v = W_rel12[(4 * 12 + (r - 17)) * 16 + h];
        else if (r < 41) v = W_rel12[(5 * 12 + (r - 29)) * 16 + h];
        else if (r < 53) {
            int rr = r - 41;
            v = W_root12[(0 * 12 + rr) * 16 + h] + W_root12[(2 * 12 + rr) * 16 + h]
              + W_root12[(4 * 12 + rr) * 16 + h] + W_root12[(5 * 12 + rr) * 16 + h];
        } else v = 0.f;
        Wc_t[idx] = v;
    }
    // data: rows 0-11 W_rel12[0]; 12-23 W_rel12[3]; 24-28 W_root5[0]+W_root5[1]
    for (int idx = t; idx < 32 * 16; idx += 256) {
        int r = idx >> 4, h = idx & 15;
        float v;
        if (r < 12)      v = W_rel12[(0 * 12 + r) * 16 + h];
        else if (r < 24) v = W_rel12[(3 * 12 + (r - 12)) * 16 + h];
        else if (r < 29) {
            int rr = r - 24;
            v = W_root5[(0 * 5 + rr) * 16 + h] + W_root5[(1 * 5 + rr) * 16 + h];
        } else v = 0.f;
        Wc_d[idx] = v;
    }
    // devices: rows 0-11 W_rel12[1]; 12-16 W_rel5[1]; 17-28 W_root12[1]+W_root12[3]
    for (int idx = t; idx < 32 * 16; idx += 256) {
        int r = idx >> 4, h = idx & 15;
        float v;
        if (r < 12)      v = W_rel12[(1 * 12 + r) * 16 + h];
        else if (r < 17) v = W_rel5[(1 * 5 + (r - 12)) * 16 + h];
        else if (r < 29) {
            int rr = r - 17;
            v = W_root12[(1 * 12 + rr) * 16 + h] + W_root12[(3 * 12 + rr) * 16 + h];
        } else v = 0.f;
        Wc_v[idx] = v;
    }
    if (t < 16) {
        bias3[t]      = b_rel[0 * 16 + t] + b_rel[3 * 16 + t] + b_rel[6 * 16 + t] + b_rel[7 * 16 + t];
        bias3[16 + t] = b_rel[1 * 16 + t] + b_rel[5 * 16 + t];
        bias3[32 + t] = b_rel[2 * 16 + t] + b_rel[4 * 16 + t];
    }
}

// ---------------------------------------------------------------------------
// Edge scatter: F[dst, colofs:colofs+FW] += x_src[src, 0:FW] (global atomics).
// Edge index array is int64 [2, E] flat: first E = src, next E = dst.
// Templated on FW: FW==12 rows are 48B (16B aligned) -> 3x global_load_b128.
// ---------------------------------------------------------------------------
template <int FW>
__global__ void __launch_bounds__(256)
scatter_edges_kernel(const float* __restrict__ xsrc,
                     const long long* __restrict__ ei, int E,
                     float* __restrict__ F, int stride, int colofs) {
    int e = blockIdx.x * blockDim.x + threadIdx.x;
    if (e >= E) return;
    long long s = ei[e];
    long long d = ei[(long long)E + e];
    float* fr = F + d * (long long)stride + colofs;
    if constexpr (FW == 12) {
        const float4* xr = (const float4*)(xsrc + s * 12);
        float4 a = xr[0], b = xr[1], c = xr[2];
        atomicAdd(&fr[0], a.x);  atomicAdd(&fr[1], a.y);
        atomicAdd(&fr[2], a.z);  atomicAdd(&fr[3], a.w);
        atomicAdd(&fr[4], b.x);  atomicAdd(&fr[5], b.y);
        atomicAdd(&fr[6], b.z);  atomicAdd(&fr[7], b.w);
        atomicAdd(&fr[8], c.x);  atomicAdd(&fr[9], c.y);
        atomicAdd(&fr[10], c.z); atomicAdd(&fr[11], c.w);
    } else {
        const float* xr = xsrc + s * FW;
#pragma unroll
        for (int k = 0; k < FW; ++k) atomicAdd(&fr[k], xr[k]);
    }
}

// ---------------------------------------------------------------------------
// Edge scatter into the 8 device rows: stage in LDS (ds_add_f32) to avoid
// hammering 96 global addresses with millions of atomics.
// ---------------------------------------------------------------------------
template <int FW>
__global__ void __launch_bounds__(256)
scatter_to_dev_kernel(const float* __restrict__ xsrc,
                      const long long* __restrict__ ei, int E,
                      float* __restrict__ Fdev, int stride, int colofs) {
    __shared__ float acc[8 * FW];
    for (int i = threadIdx.x; i < 8 * FW; i += blockDim.x) acc[i] = 0.f;
    __syncthreads();
    long long tid = (long long)blockIdx.x * blockDim.x + threadIdx.x;
    long long nth = (long long)gridDim.x * blockDim.x;
    for (long long e = tid; e < E; e += nth) {
        long long s = ei[e];
        int d = (int)ei[(long long)E + e];
        float* ar = &acc[d * FW];
        if constexpr (FW == 12) {
            const float4* xr = (const float4*)(xsrc + s * 12);
            float4 a = xr[0], b = xr[1], c = xr[2];
            atomicAdd(&ar[0], a.x);  atomicAdd(&ar[1], a.y);
            atomicAdd(&ar[2], a.z);  atomicAdd(&ar[3], a.w);
            atomicAdd(&ar[4], b.x);  atomicAdd(&ar[5], b.y);
            atomicAdd(&ar[6], b.z);  atomicAdd(&ar[7], b.w);
            atomicAdd(&ar[8], c.x);  atomicAdd(&ar[9], c.y);
            atomicAdd(&ar[10], c.z); atomicAdd(&ar[11], c.w);
        } else {
            const float* xr = xsrc + s * FW;
#pragma unroll
            for (int k = 0; k < FW; ++k) atomicAdd(&ar[k], xr[k]);
        }
    }
    __syncthreads();
    for (int i = threadIdx.x; i < 8 * FW; i += blockDim.x) {
        int d = i / FW, k = i % FW;
        atomicAdd(&Fdev[d * stride + colofs + k], acc[i]);
    }
}

// ---------------------------------------------------------------------------
// GEMM (F[M,K] @ Wc[K,16] + bias) with V_WMMA_F32_16X16X4_F32, fused
// LayerNorm (over the 16 columns, via shfl_xor butterflies within each
// 16-lane half) + leaky ReLU. One wave per 16-row tile.
// A layout: lanes 0-15 row=lane K={k0,k0+1}; lanes 16-31 row=lane-16 K={k0+2,k0+3}.
// B layout: lanes 0-15 vgpr j -> B[k0+j][lane]; lanes 16-31 -> B[k0+2+j][lane-16].
// D layout: vgpr r, lanes 0-15 -> D[r][lane]; lanes 16-31 -> D[r+8][lane-16].
// ---------------------------------------------------------------------------
__global__ void __launch_bounds__(256)
gemm_ln_act_kernel(const float* __restrict__ F, int K,
                   const float* __restrict__ Wc,
                   const float* __restrict__ bias,
                   const float* __restrict__ gamma,
                   const float* __restrict__ beta,
                   float* __restrict__ out, int mvalid, int ntiles) {
    int wave = (int)(((long long)blockIdx.x * blockDim.x + threadIdx.x) >> 5);
    int lane = threadIdx.x & 31;
    if (wave >= ntiles) return;  // wave-uniform: EXEC stays all-ones for WMMA

    int col = lane & 15;
    int kh  = (lane >> 4) << 1;  // 0 or 2
    const float* arow = F + ((long long)wave * 16 + col) * K + kh;

    float bcol = bias[col];
    v8f acc;
#pragma unroll
    for (int r = 0; r < 8; ++r) acc[r] = bcol;  // bias folded into C once

    for (int k0 = 0; k0 < K; k0 += 4) {
        v2f a = *(const v2f*)(arow + k0);
        v2f b;
        b.x = Wc[(k0 + kh + 0) * 16 + col];
        b.y = Wc[(k0 + kh + 1) * 16 + col];
        acc = __builtin_amdgcn_wmma_f32_16x16x4_f32(
            /*neg_a=*/false, a, /*neg_b=*/false, b,
            /*c_mod=*/(short)0, acc, /*reuse_a=*/false, /*reuse_b=*/false);
    }

    float g = gamma[col], be = beta[col];
    int rowbase = wave * 16 + ((lane >> 4) << 3);
#pragma unroll
    for (int r = 0; r < 8; ++r) {
        float v = acc[r];
        float s = v;
        s += __shfl_xor(s, 1, 32); s += __shfl_xor(s, 2, 32);
        s += __shfl_xor(s, 4, 32); s += __shfl_xor(s, 8, 32);
        float mu = s * 0.0625f;
        float d = v - mu;
        float q = d * d;
        q += __shfl_xor(q, 1, 32); q += __shfl_xor(q, 2, 32);
        q += __shfl_xor(q, 4, 32); q += __shfl_xor(q, 8, 32);
        float var = q * 0.0625f;
        float y = d * rsqrtf(var + 1e-5f) * g + be;
        y = (y > 0.f) ? y : 0.01f * y;
        int row = rowbase + r;
        if (row < mvalid) out[(long long)row * 16 + col] = y;
    }
}

// ---------------------------------------------------------------------------
extern "C" void kernel_launch(void* const* d_in, const int* in_sizes, int n_in,
                              void* d_out, int out_size, void* d_ws, size_t ws_size,
                              hipStream_t stream) {
    const float* x_tasks   = (const float*)d_in[0];
    const float* x_data    = (const float*)d_in[1];
    const float* x_devices = (const float*)d_in[2];
    const long long* ei_dt   = (const long long*)d_in[3];  // data->tasks
    const long long* ei_td   = (const long long*)d_in[4];  // tasks->data
    const long long* ei_tdev = (const long long*)d_in[5];  // tasks->devices
    const long long* ei_devt = (const long long*)d_in[6];  // devices->tasks
    const long long* ei_ddev = (const long long*)d_in[7];  // data->devices
    const long long* ei_devd = (const long long*)d_in[8];  // devices->data
    const long long* ei_tt_to   = (const long long*)d_in[9];
    const long long* ei_tt_from = (const long long*)d_in[10];
    const float* W_rel5   = (const float*)d_in[11];
    const float* W_rel12  = (const float*)d_in[12];
    const float* b_rel    = (const float*)d_in[13];
    const float* W_root5  = (const float*)d_in[14];
    const float* W_root12 = (const float*)d_in[15];
    const float* ln_gamma = (const float*)d_in[16];
    const float* ln_beta  = (const float*)d_in[17];

    const int N_T = in_sizes[0] / 12;
    const int N_D = in_sizes[1] / 5;
    const int N_V = in_sizes[2] / 12;
    const int E_dt   = in_sizes[3] / 2;
    const int E_td   = in_sizes[4] / 2;
    const int E_tdev = in_sizes[5] / 2;
    const int E_devt = in_sizes[6] / 2;
    const int E_ddev = in_sizes[7] / 2;
    const int E_devd = in_sizes[8] / 2;
    const int E_tt_to   = in_sizes[9] / 2;
    const int E_tt_from = in_sizes[10] / 2;

    // ---- workspace layout (floats) ----
    float* ws = (float*)d_ws;
    size_t off = 0;
    float* F_t = ws + off; off += (size_t)N_T * 56;  // tasks features [N_T,56]
    float* F_d = ws + off; off += (size_t)N_D * 32;  // data features  [N_D,32]
    float* F_v = ws + off; off += (size_t)16 * 32;   // device features, padded to 16 rows
    float* Wc_t = ws + off; off += 56 * 16;
    float* Wc_d = ws + off; off += 32 * 16;
    float* Wc_v = ws + off; off += 32 * 16;
    float* bias3 = ws + off; off += 3 * 16;

    hipMemsetAsync(d_ws, 0, off * sizeof(float), stream);

    // ---- root feature copies ----
    {
        long long tot = (long long)N_T * 12;
        copy_root_kernel<<<(unsigned)((tot + 255) / 256), 256, 0, stream>>>(
            x_tasks, 12, F_t, 56, 41, N_T);
        tot = (long long)N_D * 5;
        copy_root_kernel<<<(unsigned)((tot + 255) / 256), 256, 0, stream>>>(
            x_data, 5, F_d, 32, 24, N_D);
        copy_root_kernel<<<1, 256, 0, stream>>>(x_devices, 12, F_v, 32, 17, N_V);
    }

    // ---- combined weights + biases ----
    build_weights_kernel<<<1, 256, 0, stream>>>(W_rel5, W_rel12, b_rel, W_root5,
                                                W_root12, Wc_t, Wc_d, Wc_v, bias3);

    // ---- edge scatters (raw-feature aggregation) ----
    scatter_edges_kernel<5><<<(E_dt + 255) / 256, 256, 0, stream>>>(
        x_data, ei_dt, E_dt, F_t, 56, 0);
    scatter_edges_kernel<12><<<(E_devt + 255) / 256, 256, 0, stream>>>(
        x_devices, ei_devt, E_devt, F_t, 56, 5);
    scatter_edges_kernel<12><<<(E_tt_to + 255) / 256, 256, 0, stream>>>(
        x_tasks, ei_tt_to, E_tt_to, F_t, 56, 17);
    scatter_edges_kernel<12><<<(E_tt_from + 255) / 256, 256, 0, stream>>>(
        x_tasks, ei_tt_from, E_tt_from, F_t, 56, 29);
    scatter_edges_kernel<12><<<(E_td + 255) / 256, 256, 0, stream>>>(
        x_tasks, ei_td, E_td, F_d, 32, 0);
    scatter_edges_kernel<12><<<(E_devd + 255) / 256, 256, 0, stream>>>(
        x_devices, ei_devd, E_devd, F_d, 32, 12);
    // device destinations: LDS-staged reduction (8 rows only)
    scatter_to_dev_kernel<12><<<256, 256, 0, stream>>>(
        x_tasks, ei_tdev, E_tdev, F_v, 32, 0);
    scatter_to_dev_kernel<5><<<128, 256, 0, stream>>>(
        x_data, ei_ddev, E_ddev, F_v, 32, 12);

    // ---- WMMA GEMM + LayerNorm + leaky ReLU ----
    float* out = (float*)d_out;
    {
        int ntiles = (N_T + 15) / 16;
        int blocks = (ntiles + 7) / 8;  // 8 waves of 32 per block
        gemm_ln_act_kernel<<<blocks, 256, 0, stream>>>(
            F_t, 56, Wc_t, bias3 + 0, ln_gamma + 0, ln_beta + 0,
            out, N_T, ntiles);
    }
    {
        int ntiles = (N_D + 15) / 16;
        int blocks = (ntiles + 7) / 8;
        gemm_ln_act_kernel<<<blocks, 256, 0, stream>>>(
            F_d, 32, Wc_d, bias3 + 16, ln_gamma + 16, ln_beta + 16,
            out + (size_t)N_T * 16, N_D, ntiles);
    }
    {
        gemm_ln_act_kernel<<<1, 32, 0, stream>>>(
            F_v, 32, Wc_v, bias3 + 32, ln_gamma + 32, ln_beta + 32,
            out + ((size_t)N_T + N_D) * 16, N_V, 1);
    }
}